// RefineDetMultiBoxLoss_41497974014487
// MI455X (gfx1250) — compile-verified
//
#include <hip/hip_runtime.h>
#include <hip/hip_bf16.h>

// ---------------- problem constants ----------------
#define B_   64
#define P_   6375
#define C_   81
#define O_   16
#define NT_FULL   398          // full 16-prior tiles (398*16 = 6368)
#define TAIL_CNT  7            // 6375 - 6368
#define TAIL_OWNER 14          // 398 % 64 wave-slots
#define S2   16                // slices per batch in k_conf
#define BUFB 5248              // LDS bytes per tile buffer (>= 16*81*4 = 5184)

typedef float v2f __attribute__((ext_vector_type(2)));
typedef float v8f __attribute__((ext_vector_type(8)));
typedef unsigned int u32x4 __attribute__((ext_vector_type(4)));
typedef int i32x4 __attribute__((ext_vector_type(4)));
typedef int i32x8 __attribute__((ext_vector_type(8)));

// ---------------- CDNA5 feature plumbing ----------------
#if defined(__has_builtin)
#if __has_builtin(__builtin_amdgcn_tensor_load_to_lds)
#define HAS_TDM 1
#endif
#endif

#if __has_include(<hip/amd_detail/amd_gfx1250_TDM.h>)
#define TDM_6ARG 1
#endif

#if defined(__has_builtin) && __has_builtin(__builtin_amdgcn_s_wait_tensorcnt)
#define TWAIT(n) __builtin_amdgcn_s_wait_tensorcnt((short)(n))
#else
#define TWAIT(n) asm volatile("s_wait_tensorcnt %0" :: "n"(n))
#endif

// Fast transcendentals: lower straight to v_exp_f32 / v_log_f32 so the
// 33M-exponent stream stays well under the 132MB/23.3TB/s HBM floor.
__device__ __forceinline__ float fexp(float x) { return __expf(x); }
__device__ __forceinline__ float flog(float x) { return __logf(x); }

// Issue one TDM 2D tile load: 16 rows (priors) x 81 dwords (classes), row
// stride 81 dwords, from global byte address `ga` into LDS byte offset `lds`.
// D# layout per CDNA5 ISA 8.3-8.6.  Fallback: synchronous per-lane copy.
__device__ __forceinline__ void tile_copy(const float* __restrict__ gsrc,
                                          char* lbuf, unsigned lds_off, int lane) {
#if defined(HAS_TDM)
  unsigned long long ga = (unsigned long long)(size_t)gsrc;
  u32x4 g0;
  g0[0] = 1u;                                   // count=1 (valid), no gather
  g0[1] = lds_off;                              // lds_addr (bytes)
  g0[2] = (unsigned)(ga & 0xFFFFFFFFull);       // global_addr[31:0]
  g0[3] = (unsigned)(ga >> 32) | (2u << 30);    // global_addr[56:32] | type=2
  i32x8 g1;
  g1[0] = 0x00020000;                           // wg_mask=0, data_size=2 (4B)
  g1[1] = (int)(81u << 16);                     // tensor_dim0 = 81 (bits 79:48)
  g1[2] = 0;                                    // tensor_dim1[15:0] in [31:16] = 0
  g1[3] = (int)((81u << 16) | 1u);              // tensor_dim1 hi16=1 (dim1=65536), tile_dim0=81
  g1[4] = 16;                                   // tile_dim1 = 16, tile_dim2 = 0
  g1[5] = 81;                                   // tensor_dim0_stride = 81
  g1[6] = 0;
  g1[7] = 0;
  i32x4 gz; gz[0]=0; gz[1]=0; gz[2]=0; gz[3]=0;
#if defined(TDM_6ARG)
  i32x8 gz8; gz8[0]=0; gz8[1]=0; gz8[2]=0; gz8[3]=0; gz8[4]=0; gz8[5]=0; gz8[6]=0; gz8[7]=0;
  __builtin_amdgcn_tensor_load_to_lds(g0, g1, gz, gz, gz8, 0);
#else
  __builtin_amdgcn_tensor_load_to_lds(g0, g1, gz, gz, 0);
#endif
#else
  (void)lds_off;
  float* lf = (float*)lbuf;
  for (int idx = lane; idx < 16 * 81; idx += 32) lf[idx] = gsrc[idx];
#endif
}

__device__ __forceinline__ float sl1(float d) {
  float ad = fabsf(d);
  return (ad < 1.0f) ? 0.5f * d * d : ad - 0.5f;
}

// ============================================================
// Kernel 1: matching, conf_t, SmoothL1 loc loss, num_pos
// ============================================================
__global__ void __launch_bounds__(256)
k_match(const float* __restrict__ arm_loc, const float* __restrict__ priors,
        const float* __restrict__ tboxes, const int* __restrict__ tlabels,
        int* __restrict__ conft, int* __restrict__ numpos, float* __restrict__ lossl) {
  __shared__ float s_bto[P_];
  __shared__ int   s_bti[P_];
  __shared__ unsigned long long s_bpk[O_];
  __shared__ float s_tr[O_ * 4];
  __shared__ float s_ga[O_];
  __shared__ int   s_lb[O_];
  __shared__ float s_redf[256];
  __shared__ int   s_redi[256];

  const int b = blockIdx.x, tid = threadIdx.x;
  if (tid < O_ * 4) s_tr[tid] = tboxes[b * O_ * 4 + tid];
  if (tid < O_) { s_lb[tid] = tlabels[b * O_ + tid]; s_bpk[tid] = 0ull; }
  __syncthreads();
  if (tid < O_) {
    float x1 = s_tr[tid*4], y1 = s_tr[tid*4+1], x2 = s_tr[tid*4+2], y2 = s_tr[tid*4+3];
    s_ga[tid] = (x2 - x1) * (y2 - y1);
  }
  __syncthreads();

  unsigned long long lb[O_];
#pragma unroll
  for (int g = 0; g < O_; ++g) lb[g] = 0ull;

  const float4* pr4 = (const float4*)priors;
  for (int p = tid; p < P_; p += 256) {
    float4 pv = pr4[p];
    float px1 = pv.x - pv.z * 0.5f, py1 = pv.y - pv.w * 0.5f;
    float px2 = pv.x + pv.z * 0.5f, py2 = pv.y + pv.w * 0.5f;
    float pa = (px2 - px1) * (py2 - py1);
    float bv = -1.0f; int bi = 0;
#pragma unroll
    for (int g = 0; g < O_; ++g) {
      float ix1 = fmaxf(px1, s_tr[g*4+0]), iy1 = fmaxf(py1, s_tr[g*4+1]);
      float ix2 = fminf(px2, s_tr[g*4+2]), iy2 = fminf(py2, s_tr[g*4+3]);
      float iw = fmaxf(ix2 - ix1, 0.0f), ih = fmaxf(iy2 - iy1, 0.0f);
      float inter = iw * ih;
      float iou = inter / (s_ga[g] + pa - inter);
      if (iou > bv) { bv = iou; bi = g; }                       // first-max tie rule
      unsigned long long key = ((unsigned long long)__float_as_uint(iou) << 32)
                             | (unsigned long long)(0xFFFFFFFFu - (unsigned)p);
      lb[g] = key > lb[g] ? key : lb[g];
    }
    s_bto[p] = bv; s_bti[p] = bi;
  }
#pragma unroll
  for (int g = 0; g < O_; ++g) atomicMax(&s_bpk[g], lb[g]);
  __syncthreads();
  if (tid == 0) {
    for (int g = 0; g < O_; ++g) {                               // later g wins
      int p = (int)(0xFFFFFFFFu - (unsigned)(s_bpk[g] & 0xFFFFFFFFull));
      s_bto[p] = 2.0f; s_bti[p] = g;
    }
  }
  __syncthreads();

  int np = 0; float ll = 0.0f;
  const float4* al4 = (const float4*)arm_loc;
  for (int p = tid; p < P_; p += 256) {
    int g = s_bti[p];
    int ct = (s_bto[p] < 0.5f) ? 0 : s_lb[g];
    conft[(size_t)b * P_ + p] = ct;
    if (ct > 0) {
      ++np;
      float4 pv = pr4[p];
      float mx1 = s_tr[g*4+0], my1 = s_tr[g*4+1], mx2 = s_tr[g*4+2], my2 = s_tr[g*4+3];
      float gcx = ((mx1 + mx2) * 0.5f - pv.x) / (0.1f * pv.z);
      float gcy = ((my1 + my2) * 0.5f - pv.y) / (0.1f * pv.w);
      float gw  = flog((mx2 - mx1) / pv.z) / 0.2f;
      float gh  = flog((my2 - my1) / pv.w) / 0.2f;
      float4 ld = al4[(size_t)b * P_ + p];
      ll += sl1(ld.x - gcx) + sl1(ld.y - gcy) + sl1(ld.z - gw) + sl1(ld.w - gh);
    }
  }
  s_redf[tid] = ll; s_redi[tid] = np;
  __syncthreads();
  for (int s = 128; s > 0; s >>= 1) {
    if (tid < s) { s_redf[tid] += s_redf[tid + s]; s_redi[tid] += s_redi[tid + s]; }
    __syncthreads();
  }
  if (tid == 0) { lossl[b] = s_redf[0]; numpos[b] = s_redi[0]; }
}

// ============================================================
// Kernel 2: per-prior logsumexp via TDM tiles + WMMA f32 16x16x4
// writes mining score array (0 for positives) + pos-CE partials
// ============================================================
__global__ void __launch_bounds__(128)
k_conf(const float* __restrict__ conf, const int* __restrict__ conft,
       float* __restrict__ lossrank, float* __restrict__ pcpart) {
  __shared__ __align__(16) char smem[4 * 2 * BUFB];
  const int bidx  = blockIdx.x / S2;
  const int slice = blockIdx.x % S2;
  const int wid  = threadIdx.x >> 5;
  const int lane = threadIdx.x & 31;
  const int wslot = slice * 4 + wid;
  const int m = lane & 15, half = lane >> 4;

  char* bufs = smem + wid * 2 * BUFB;
  const unsigned lds0 = (unsigned)(size_t)bufs;   // LDS byte offset (low 32 of flat)
  float pc = 0.0f;
  v2f bones; bones.x = 1.0f; bones.y = 1.0f;

  int t = wslot, par = 0;
  if (t < NT_FULL)
    tile_copy(conf + ((size_t)bidx * P_ + (size_t)t * 16) * C_, bufs, lds0, lane);

  for (; t < NT_FULL; t += 64) {
    int tn = t + 64;
    if (tn < NT_FULL) {
      tile_copy(conf + ((size_t)bidx * P_ + (size_t)tn * 16) * C_,
                bufs + (par ^ 1) * BUFB, lds0 + (unsigned)((par ^ 1) * BUFB), lane);
      TWAIT(1);                                  // current tile's TDM done
    } else {
      TWAIT(0);
    }
    asm volatile("" ::: "memory");
    const float* tf = (const float*)(bufs + par * BUFB);

    // phase A: per-prior max (lane covers classes 4k+2*half, +1)
    float mx = -3.4e38f;
#pragma unroll
    for (int k = 0; k < 21; ++k) {
      int c0 = 4 * k + 2 * half;
      if (c0 < C_)     mx = fmaxf(mx, tf[m * C_ + c0]);
      if (c0 + 1 < C_) mx = fmaxf(mx, tf[m * C_ + c0 + 1]);
    }
    mx = fmaxf(mx, __shfl_xor(mx, 16, 32));
    float pm[8];
#pragma unroll
    for (int r = 0; r < 8; ++r) pm[r] = __shfl(mx, half * 8 + r, 32);

    // phase B: exp then WMMA-accumulated row sums (B = ones)
    v8f acc = {0.f, 0.f, 0.f, 0.f, 0.f, 0.f, 0.f, 0.f};
#pragma unroll
    for (int k = 0; k < 21; ++k) {
      int c0 = 4 * k + 2 * half;
      float a0 = (c0 < C_)     ? fexp(tf[m * C_ + c0] - mx)     : 0.0f;
      float a1 = (c0 + 1 < C_) ? fexp(tf[m * C_ + c0 + 1] - mx) : 0.0f;
      v2f a; a.x = a0; a.y = a1;
      acc = __builtin_amdgcn_wmma_f32_16x16x4_f32(false, a, false, bones,
                                                  (short)0, acc, false, false);
    }
    // lanes 0 / 16 hold rows 0..7 / 8..15 in acc[0..7]
    if (m == 0) {
#pragma unroll
      for (int r = 0; r < 8; ++r) {
        int pl = half * 8 + r;
        size_t gp = (size_t)bidx * P_ + (size_t)(t * 16 + pl);
        float lse = flog(acc[r]) + pm[r];
        int ct = conft[gp];
        float v = lse - tf[pl * C_ + ct];
        if (ct > 0) { pc += v; lossrank[gp] = 0.0f; }
        else        { lossrank[gp] = v; }
      }
    }
    par ^= 1;
  }

  // tail tile (7 priors): plain scalar path
  if (TAIL_CNT > 0 && wslot == TAIL_OWNER) {
    if (lane < TAIL_CNT) {
      int gpi = NT_FULL * 16 + lane;
      const float* row = conf + ((size_t)bidx * P_ + gpi) * C_;
      float mx = -3.4e38f;
      for (int c = 0; c < C_; ++c) mx = fmaxf(mx, row[c]);
      float se = 0.0f;
      for (int c = 0; c < C_; ++c) se += fexp(row[c] - mx);
      float lse = flog(se) + mx;
      size_t gp = (size_t)bidx * P_ + gpi;
      int ct = conft[gp];
      float v = lse - row[ct];
      if (ct > 0) { pc += v; lossrank[gp] = 0.0f; }
      else        { lossrank[gp] = v; }
    }
  }

  for (int off = 16; off > 0; off >>= 1) pc += __shfl_xor(pc, off, 32);
  if (lane == 0) pcpart[bidx * 64 + wslot] = pc;
}

// ============================================================
// Kernel 3: per-batch top-k sum via LDS bitonic sort (descending)
// ============================================================
#define N2 8192
__global__ void __launch_bounds__(512)
k_topk(const float* __restrict__ lossrank, const int* __restrict__ numpos,
       float* __restrict__ negsum) {
  __shared__ float sv[N2];
  __shared__ float sred[512];
  const int b = blockIdx.x, tid = threadIdx.x;
  for (int i = tid; i < N2; i += 512)
    sv[i] = (i < P_) ? lossrank[(size_t)b * P_ + i] : -1.0f;
  __syncthreads();
  for (int k2 = 2; k2 <= N2; k2 <<= 1) {
    for (int j = k2 >> 1; j > 0; j >>= 1) {
      for (int i = tid; i < N2; i += 512) {
        int ij = i ^ j;
        if (ij > i) {
          float a = sv[i], c = sv[ij];
          bool desc = ((i & k2) == 0);
          if (desc ? (a < c) : (a > c)) { sv[i] = c; sv[ij] = a; }
        }
      }
      __syncthreads();
    }
  }
  int np = numpos[b];
  int k = 3 * np; if (k > P_ - 1) k = P_ - 1; if (k < 0) k = 0;
  float s = 0.0f;
  for (int i = tid; i < k; i += 512) s += sv[i];
  sred[tid] = s;
  __syncthreads();
  for (int st = 256; st > 0; st >>= 1) {
    if (tid < st) sred[tid] += sred[tid + st];
    __syncthreads();
  }
  if (tid == 0) negsum[b] = sred[0];
}

// ============================================================
// Kernel 4: final reduction to the two scalars
// ============================================================
__global__ void __launch_bounds__(64)
k_final(const int* __restrict__ numpos, const float* __restrict__ lossl,
        const float* __restrict__ pcpart, const float* __restrict__ negsum,
        float* __restrict__ out) {
  __shared__ float sl[64], sc[64];
  __shared__ int sn[64];
  const int t = threadIdx.x;
  float pcsum = 0.0f;
  for (int w = 0; w < 64; ++w) pcsum += pcpart[t * 64 + w];
  sl[t] = lossl[t]; sc[t] = pcsum + negsum[t]; sn[t] = numpos[t];
  __syncthreads();
  for (int s = 32; s > 0; s >>= 1) {
    if (t < s) { sl[t] += sl[t + s]; sc[t] += sc[t + s]; sn[t] += sn[t + s]; }
    __syncthreads();
  }
  if (t == 0) {
    float N = (float)sn[0];
    if (N < 1.0f) N = 1.0f;
    out[0] = sl[0] / N;
    out[1] = sc[0] / N;
  }
}

// ============================================================
extern "C" void kernel_launch(void* const* d_in, const int* in_sizes, int n_in,
                              void* d_out, int out_size, void* d_ws, size_t ws_size,
                              hipStream_t stream) {
  (void)in_sizes; (void)n_in; (void)out_size; (void)ws_size;
  const float* arm_loc  = (const float*)d_in[0];
  const float* arm_conf = (const float*)d_in[1];
  // d_in[2], d_in[3]: odm_* unused (use_ARM=False path)
  const float* priors   = (const float*)d_in[4];
  const float* tboxes   = (const float*)d_in[5];
  const int*   tlabels  = (const int*)d_in[6];

  const size_t BP = (size_t)B_ * P_;
  char* w = (char*)d_ws;
  float* ws_lossrank = (float*)w;                         // B*P floats
  int*   ws_conft    = (int*)(w + BP * 4);                // B*P ints
  int*   ws_numpos   = (int*)(w + 2 * BP * 4);            // B ints
  float* ws_lossl    = (float*)(w + 2 * BP * 4 + 256);    // B floats
  float* ws_pcpart   = (float*)(w + 2 * BP * 4 + 512);    // B*64 floats
  float* ws_negsum   = (float*)(w + 2 * BP * 4 + 512 + B_ * 64 * 4); // B floats

  k_match<<<B_, 256, 0, stream>>>(arm_loc, priors, tboxes, tlabels,
                                  ws_conft, ws_numpos, ws_lossl);
  k_conf<<<B_ * S2, 128, 0, stream>>>(arm_conf, ws_conft, ws_lossrank, ws_pcpart);
  k_topk<<<B_, 512, 0, stream>>>(ws_lossrank, ws_numpos, ws_negsum);
  k_final<<<1, 64, 0, stream>>>(ws_numpos, ws_lossl, ws_pcpart, ws_negsum,
                                (float*)d_out);
}